// KeypointSampler_38001870635222
// MI455X (gfx1250) — compile-verified
//
#include <hip/hip_runtime.h>
#include <stdint.h>
#include <math.h>

// ---------------------------------------------------------------------------
// KeypointSampler for MI455X (gfx1250, wave32).
// Memory-bound (143 MB @ 23.3 TB/s ~ 6.1 us). No matrix ops -> no WMMA; the
// CDNA5 feature used is the async global->LDS copy path (ASYNCcnt) to
// gather/transpose row-major image data into window-major LDS tiles.
// ---------------------------------------------------------------------------

#define WSZ 8
#define IMG 2048
#define NW  256                    // windows per image dimension (2048/8)
#define WIN_STRIDE 68              // dwords per window slot in LDS (64 + 4 pad, 16B aligned)
#define TILE_DW (NW * WIN_STRIDE)  // 17408 dwords per staged window-row
#define LDS_BYTES (2 * TILE_DW * 4)  // 139264 bytes (two tiles: batch b and b+4)

#define CAT_HALF  16777216u        // n/2 for the 8*1*256*256*64 categorical uniforms
#define BERN_HALF 262144u          // n/2 for the 8*1*256*256 bernoulli uniforms

typedef int v4i __attribute__((ext_vector_type(4)));

// ---- JAX threefry2x32 (20 rounds, 5 key injections), host+device ----------
__host__ __device__ static inline uint32_t rotl32(uint32_t x, int r) {
  return (x << r) | (x >> (32 - r));
}

__host__ __device__ static inline void threefry2x32(
    uint32_t k0, uint32_t k1, uint32_t x0, uint32_t x1,
    uint32_t& o0, uint32_t& o1) {
  const uint32_t k2 = k0 ^ k1 ^ 0x1BD11BDAu;
  x0 += k0; x1 += k1;
#define TF_R(r) { x0 += x1; x1 = rotl32(x1, (r)); x1 ^= x0; }
  TF_R(13) TF_R(15) TF_R(26) TF_R(6)  x0 += k1; x1 += k2 + 1u;
  TF_R(17) TF_R(29) TF_R(16) TF_R(24) x0 += k2; x1 += k0 + 2u;
  TF_R(13) TF_R(15) TF_R(26) TF_R(6)  x0 += k0; x1 += k1 + 3u;
  TF_R(17) TF_R(29) TF_R(16) TF_R(24) x0 += k1; x1 += k2 + 4u;
  TF_R(13) TF_R(15) TF_R(26) TF_R(6)  x0 += k2; x1 += k0 + 5u;
#undef TF_R
  o0 = x0; o1 = x1;
}

// ---- device helpers -------------------------------------------------------
__device__ __forceinline__ float bits_to_u01(uint32_t b) {
  // JAX: bitcast((bits >> 9) | 0x3f800000) - 1.0  ->  [0, 1)
  return __uint_as_float((b >> 9) | 0x3f800000u) - 1.0f;
}

__device__ __forceinline__ float logsigf(float v) {
  // jax.nn.log_sigmoid(x) = -(max(-x,0) + log1p(exp(-|x|)))
  const float l = log1pf(__expf(-fabsf(v)));
  return (v >= 0.f) ? -l : v - l;
}

// CDNA5 async copy: 16 bytes global -> LDS, tracked by ASYNCcnt.
__device__ __forceinline__ void async_b128_to_lds(const float* g, const float* l) {
#if __has_builtin(__builtin_amdgcn_global_load_async_to_lds_b128)
  __builtin_amdgcn_global_load_async_to_lds_b128(
      (__attribute__((address_space(1))) v4i*)(uintptr_t)g,
      (__attribute__((address_space(3))) v4i*)(uint32_t)(uintptr_t)l,
      0, 0);
#else
  const uint32_t laddr = (uint32_t)(uintptr_t)l;  // LDS byte address
  asm volatile("global_load_async_to_lds_b128 %0, %1, off"
               :: "v"(laddr), "v"(g) : "memory");
#endif
}

__device__ __forceinline__ void wait_async_zero() {
#if __has_builtin(__builtin_amdgcn_s_wait_asynccnt)
  __builtin_amdgcn_s_wait_asynccnt(0);
#else
  asm volatile("s_wait_asynccnt 0" ::: "memory");
#endif
}

// ---------------------------------------------------------------------------
__global__ __launch_bounds__(256) void keypoint_sampler_kernel(
    const float* __restrict__ x, float* __restrict__ out,
    uint32_t kc0, uint32_t kc1, uint32_t kb0, uint32_t kb1) {
  extern __shared__ float lds[];
  const int tid = threadIdx.x;          // window x index (0..255)
  const int b   = blockIdx.x >> 8;      // 0..3  (batch; pairs with b+4)
  const int wy  = blockIdx.x & 255;     // window y index

  // ---- Stage two window-rows (batch b and b+4) into LDS, window-major -----
  #pragma unroll
  for (int half = 0; half < 2; ++half) {
    const float* gbase = x + ((size_t)(b + 4 * half) * IMG + (size_t)wy * WSZ) * (size_t)IMG;
    const int tb = half * TILE_DW;
    #pragma unroll
    for (int i = 0; i < 16; ++i) {
      const int chunk = i * 256 + tid;  // 0..4095 : 16B chunks of a 64KB tile
      const int r  = chunk >> 9;        // image row within the window-row (0..7)
      const int cj = chunk & 511;       // 16B chunk index within the 2048-float row
      const int wx = cj >> 1;           // destination window
      const int hf = cj & 1;            // low/high half of the window row
      const float* g = gbase + r * IMG + cj * 4;
      const float* l = lds + tb + wx * WIN_STRIDE + r * 8 + hf * 4;
      async_b128_to_lds(g, l);
    }
  }
  wait_async_zero();
  __syncthreads();

  const uint32_t wA = ((uint32_t)b * NW + (uint32_t)wy) * NW + (uint32_t)tid;
  const float4* pA = (const float4*)(lds + tid * WIN_STRIDE);
  const float4* pB = (const float4*)(lds + TILE_DW + tid * WIN_STRIDE);
  const float TINY = 1.17549435e-38f;

  // ---- pass 1: max + Gumbel argmax. One threefry pair feeds BOTH windows. --
  float mA = -INFINITY, mB = -INFINITY;
  float bestA = -INFINITY, bestB = -INFINITY;
  float selA = 0.f, selB = 0.f;
  int cA = 0, cB = 0;
  const uint32_t cbase = wA * 64u;
  for (int kb = 0; kb < 16; ++kb) {
    const float4 qA = pA[kb];
    const float4 qB = pB[kb];
    #pragma unroll
    for (int j = 0; j < 4; ++j) {
      const int k = kb * 4 + j;
      const float xA = (j == 0) ? qA.x : (j == 1) ? qA.y : (j == 2) ? qA.z : qA.w;
      const float xB = (j == 0) ? qB.x : (j == 1) ? qB.y : (j == 2) ? qB.z : qB.w;
      uint32_t r0, r1;
      threefry2x32(kc0, kc1, cbase + (uint32_t)k, cbase + (uint32_t)k + CAT_HALF, r0, r1);
      const float gA = -__logf(-__logf(fmaxf(bits_to_u01(r0), TINY)));
      const float gB = -__logf(-__logf(fmaxf(bits_to_u01(r1), TINY)));
      const float scA = xA + gA;
      if (scA > bestA) { bestA = scA; selA = xA; cA = k; }
      const float scB = xB + gB;
      if (scB > bestB) { bestB = scB; selB = xB; cB = k; }
      mA = fmaxf(mA, xA);
      mB = fmaxf(mB, xB);
    }
  }

  // ---- pass 2: sum exp(x - m) ---------------------------------------------
  float sA = 0.f, sB = 0.f;
  for (int kb = 0; kb < 16; ++kb) {
    const float4 qA = pA[kb];
    const float4 qB = pB[kb];
    sA += __expf(qA.x - mA) + __expf(qA.y - mA) + __expf(qA.z - mA) + __expf(qA.w - mA);
    sB += __expf(qB.x - mB) + __expf(qB.y - mB) + __expf(qB.z - mB) + __expf(qB.w - mB);
  }
  const float lseA = mA + __logf(sA);
  const float lseB = mB + __logf(sB);

  // ---- Bernoulli accept (one threefry pair feeds both windows) ------------
  uint32_t r0, r1;
  threefry2x32(kb0, kb1, wA, wA + BERN_HALF, r0, r1);
  const int accA = bits_to_u01(r0) < 1.f / (1.f + __expf(-selA));
  const int accB = bits_to_u01(r1) < 1.f / (1.f + __expf(-selB));
  const float lpA = (selA - lseA) + (accA ? logsigf(selA) : logsigf(-selA));
  const float lpB = (selB - lseB) + (accB ? logsigf(selB) : logsigf(-selB));

  // ---- write outputs: kp (B,256,256,2) [x=col,y=row], logp, mask ----------
  const uint32_t LP_OFF = 1048576u;   // 8*256*256*2
  const uint32_t MK_OFF = 1572864u;   // + 8*256*256
  const uint32_t wB = wA + BERN_HALF; // same window, batch b+4
  float2* kp = (float2*)out;
  kp[wA] = make_float2((float)(tid * WSZ + (cA & 7)), (float)(wy * WSZ + (cA >> 3)));
  kp[wB] = make_float2((float)(tid * WSZ + (cB & 7)), (float)(wy * WSZ + (cB >> 3)));
  out[LP_OFF + wA] = lpA;
  out[LP_OFF + wB] = lpB;
  out[MK_OFF + wA] = accA ? 1.0f : 0.0f;
  out[MK_OFF + wB] = accB ? 1.0f : 0.0f;
}

// ---------------------------------------------------------------------------
extern "C" void kernel_launch(void* const* d_in, const int* in_sizes, int n_in,
                              void* d_out, int out_size, void* d_ws, size_t ws_size,
                              hipStream_t stream) {
  (void)in_sizes; (void)n_in; (void)out_size; (void)d_ws; (void)ws_size;
  const float* x = (const float*)d_in[0];
  float* out = (float*)d_out;

  // jax.random.split(jax.random.key(42)):
  //   counts [0,1,2,3] -> pairs (0,2) and (1,3); k_cat = row0, k_bern = row1.
  uint32_t a0, b0, a1, b1;
  threefry2x32(0u, 42u, 0u, 2u, a0, b0);
  threefry2x32(0u, 42u, 1u, 3u, a1, b1);
  const uint32_t kc0 = a0, kc1 = a1;   // k_cat
  const uint32_t kb0 = b0, kb1 = b1;   // k_bern

  keypoint_sampler_kernel<<<dim3(1024), dim3(256), LDS_BYTES, stream>>>(
      x, out, kc0, kc1, kb0, kb1);
}